// MLA_62483184222772
// MI455X (gfx1250) — compile-verified
//
#include <hip/hip_runtime.h>
#include <hip/hip_bf16.h>
#include <math.h>

typedef float v2f __attribute__((ext_vector_type(2)));
typedef float v4f __attribute__((ext_vector_type(4)));
typedef float v8f __attribute__((ext_vector_type(8)));

// D = A(16x4) * B(4x16) + C  -- fp32 WMMA, wave32
#define WMMA_F32(a, b, c) \
  __builtin_amdgcn_wmma_f32_16x16x4_f32(false, (a), false, (b), (short)0, (c), false, false)

// ---------------------------------------------------------------------------
// Tiled GEMM: out[m, map(n)] = sum_k A[m,k]*W[k,n] + bias[n]
// map(n) = (n/group)*head_stride + col_off + n%group   (group=N -> plain)
// Block: 128 threads = 4 waves; block tile 128x64; K step 32.
// Each wave computes a 32x64 strip (2 A fragments x 4 B fragments).
// W tile stored pair-packed in LDS: Bs[(k>>1)*128 + n*2 + (k&1)] so a WMMA
// B fragment is one aligned ds_load_b64 (no VGPR repacking movs).
// Requires M%128==0, N%64==0, K%32==0.
// ---------------------------------------------------------------------------
__global__ __launch_bounds__(128)
void mla_gemm_bias_kernel(const float* __restrict__ A, const float* __restrict__ W,
                          const float* __restrict__ bias, float* __restrict__ out,
                          int M, int K, int N, int ldo,
                          int group, int head_stride, int col_off)
{
    __shared__ float As[128 * 32];      // 16 KB
    __shared__ float Bs[16 * 64 * 2];   // 8 KB, pair-packed
    const int tid  = threadIdx.x;
    const int w    = tid >> 5;
    const int lane = tid & 31;
    const int nl   = lane & 15;
    const int hi   = lane >> 4;
    const int khi  = hi * 2;
    const int m0   = blockIdx.x * 128;
    const int n0   = blockIdx.y * 64;

    v8f acc[2][4];
    for (int j = 0; j < 4; ++j) {
        float bv = bias[n0 + j * 16 + nl];
        for (int r = 0; r < 8; ++r) { acc[0][j][r] = bv; acc[1][j][r] = bv; }
    }

    for (int k0 = 0; k0 < K; k0 += 32) {
        __syncthreads();
        // A tile 128x32 (1024 float4, 8 per thread)
        for (int i = tid; i < 1024; i += 128) {
            int r = i >> 3, c = (i & 7) << 2;
            *(v4f*)&As[r * 32 + c] = *(const v4f*)&A[(size_t)(m0 + r) * K + k0 + c];
        }
        // W tile 32x64, scattered pair-packed
        for (int i = tid; i < 512; i += 128) {
            int r = i >> 4, cc = (i & 15) << 2;
            v4f wv = *(const v4f*)&W[(size_t)(k0 + r) * N + n0 + cc];
            int base = (r >> 1) * 128 + cc * 2 + (r & 1);
            Bs[base]     = wv[0];
            Bs[base + 2] = wv[1];
            Bs[base + 4] = wv[2];
            Bs[base + 6] = wv[3];
        }
        if (k0 + 32 < K) {  // hint next A tile toward the caches
            __builtin_prefetch(&A[(size_t)(m0 + tid) * K + k0 + 32], 0, 1);
        }
        __syncthreads();

        const int ar0 = w * 32 + nl;
        for (int c = 0; c < 8; ++c) {
            int kk = c * 4 + khi;                 // even
            v2f a0 = *(const v2f*)&As[ar0 * 32 + kk];
            v2f a1 = *(const v2f*)&As[(ar0 + 16) * 32 + kk];
            const float* bb = &Bs[(kk >> 1) * 128];
            for (int j = 0; j < 4; ++j) {
                v2f b = *(const v2f*)&bb[(j * 16 + nl) * 2];
                acc[0][j] = WMMA_F32(a0, b, acc[0][j]);
                acc[1][j] = WMMA_F32(a1, b, acc[1][j]);
            }
        }
    }

    for (int u = 0; u < 2; ++u) {
        for (int j = 0; j < 4; ++j) {
            int col = n0 + j * 16 + nl;
            int g = col / group;
            int d = col - g * group;
            size_t cidx = (size_t)g * head_stride + col_off + d;
            for (int r = 0; r < 8; ++r) {
                int row = m0 + w * 32 + u * 16 + r + hi * 8;
                out[(size_t)row * ldo + cidx] = acc[u][j][r];
            }
        }
    }
}

// ---------------------------------------------------------------------------
// RoPE (reference rotates only the first 16 of the 32 rope dims; the last 16
// pass through). buf layout (B*S, H, 128), rope dims at 96..127.
// One thread per (m, h). Optionally writes rotated 32-dim block to rot_out.
// ---------------------------------------------------------------------------
__global__ void mla_rope_kernel(float* __restrict__ buf, float* __restrict__ rot_out,
                                int total, int S)
{
    int idx = blockIdx.x * blockDim.x + threadIdx.x;
    if (idx >= total) return;
    const int m = idx >> 4;      // H == 16
    const int h = idx & 15;
    const float t = (float)(m % S) / 40.0f;
    // inv_freq[i] = 10000^(-i/8)
    const float inv_freq[8] = {1.0f, 0.316227766f, 0.1f, 0.0316227766f,
                               0.01f, 3.16227766e-3f, 1.0e-3f, 3.16227766e-4f};
    float* p = buf + (size_t)m * 2048 + h * 128 + 96;
    float x[32];
    for (int i = 0; i < 32; ++i) x[i] = p[i];
    float y[16];
    for (int i = 0; i < 8; ++i) {
        float f = t * inv_freq[i];
        float c = cosf(f), s = sinf(f);
        y[i]     = x[i] * c - x[i + 8] * s;
        y[i + 8] = x[i + 8] * c + x[i] * s;
    }
    for (int i = 0; i < 16; ++i) p[i] = y[i];
    if (rot_out) {
        float* q = rot_out + (size_t)idx * 32;
        for (int i = 0; i < 16; ++i) q[i] = y[i];
        for (int i = 16; i < 32; ++i) q[i] = x[i];
    }
}

// ---------------------------------------------------------------------------
// Causal attention, flash style. Block = 64 threads (2 waves), 32 query rows
// per block for one (b,h). Key tiles of 32 staged in LDS. Q in registers
// (A layout, pre-scaled). Online softmax; P transposed through LDS for P*V.
// V tile stored pair-packed so P*V B fragments are single ds_load_b64.
// Q/K/V layout: (B*S, H, 128) i.e. row stride 2048, head offset h*128.
// O layout: (B, S, H, 128).
// ---------------------------------------------------------------------------
__global__ __launch_bounds__(64)
void mla_attn_kernel(const float* __restrict__ Q, const float* __restrict__ Kf,
                     const float* __restrict__ V, float* __restrict__ O, int Sl)
{
    __shared__ float Ks[32 * 128];      // row-major (QK^T B frags contiguous)
    __shared__ float Vs[16 * 128 * 2];  // pair-packed: [(k>>1)*256 + d*2 + (k&1)]
    __shared__ float Ps[2 * 16 * 32];
    const int tid  = threadIdx.x;
    const int w    = tid >> 5;
    const int lane = tid & 31;
    const int nl   = lane & 15;
    const int hi   = lane >> 4;
    const int khi  = hi * 2;
    const int q0   = blockIdx.x * 32;
    const int h    = blockIdx.y;
    const int b    = blockIdx.z;
    const size_t bh_base = (size_t)b * Sl * 2048 + (size_t)h * 128;

    // Q rows into A-layout registers, pre-scaled by 1/sqrt(D_HEAD)
    const float scale = 0.08838834764831845f;  // 1/sqrt(128)
    const int qrow = q0 + w * 16 + nl;
    v2f qa[32];
    {
        const float* qp = Q + bh_base + (size_t)qrow * 2048;
        for (int c = 0; c < 32; ++c) {
            v2f qv = *(const v2f*)&qp[c * 4 + khi];
            qa[c][0] = qv[0] * scale;
            qa[c][1] = qv[1] * scale;
        }
    }

    v8f o[8];
    for (int t = 0; t < 8; ++t)
        for (int r = 0; r < 8; ++r) o[t][r] = 0.0f;
    float mrow[8], lrow[8];
    for (int r = 0; r < 8; ++r) { mrow[r] = -INFINITY; lrow[r] = 0.0f; }

    const int diag = q0 / 32;
    for (int kt = 0; kt <= diag; ++kt) {
        __syncthreads();
        // stage K (row-major) and V (pair-packed) 32x128 tiles
        for (int i = tid; i < 1024; i += 64) {
            int r = i >> 5, c = (i & 31) << 2;
            size_t g = bh_base + (size_t)(kt * 32 + r) * 2048 + c;
            *(v4f*)&Ks[r * 128 + c] = *(const v4f*)&Kf[g];
            v4f vv = *(const v4f*)&V[g];
            int vb = (r >> 1) * 256 + c * 2 + (r & 1);
            Vs[vb]     = vv[0];
            Vs[vb + 2] = vv[1];
            Vs[vb + 4] = vv[2];
            Vs[vb + 6] = vv[3];
        }
        __syncthreads();

        // scores: two 16x16 tiles per wave (16 rows x 32 keys)
        v8f s[2];
        for (int j = 0; j < 2; ++j)
            for (int r = 0; r < 8; ++r) s[j][r] = 0.0f;
        for (int c = 0; c < 32; ++c) {
            int kk = c * 4 + khi;
            for (int j = 0; j < 2; ++j) {
                v2f bk = *(const v2f*)&Ks[(j * 16 + nl) * 128 + kk];
                s[j] = WMMA_F32(qa[c], bk, s[j]);
            }
        }

        if (kt == diag) {  // causal mask, diagonal tile only
            for (int j = 0; j < 2; ++j) {
                int col = kt * 32 + j * 16 + nl;
                for (int r = 0; r < 8; ++r) {
                    int row = q0 + w * 16 + r + hi * 8;
                    if (col > row) s[j][r] = -INFINITY;
                }
            }
        }

        // online softmax (row reductions across the 16-lane group)
        float p0[8], p1[8], alpha[8];
        for (int r = 0; r < 8; ++r) {
            float mx = fmaxf(s[0][r], s[1][r]);
            for (int msk = 1; msk <= 8; msk <<= 1)
                mx = fmaxf(mx, __shfl_xor(mx, msk, 32));
            float mn = fmaxf(mrow[r], mx);
            alpha[r] = expf(mrow[r] - mn);
            mrow[r] = mn;
            p0[r] = expf(s[0][r] - mn);
            p1[r] = expf(s[1][r] - mn);
            float rs = p0[r] + p1[r];
            for (int msk = 1; msk <= 8; msk <<= 1)
                rs += __shfl_xor(rs, msk, 32);
            lrow[r] = lrow[r] * alpha[r] + rs;
        }
        for (int t = 0; t < 8; ++t)
            for (int r = 0; r < 8; ++r) o[t][r] *= alpha[r];

        // transpose P through per-wave LDS region into A layout
        float* pw = Ps + w * 16 * 32;
        for (int r = 0; r < 8; ++r) {
            pw[(r + hi * 8) * 32 + nl]      = p0[r];
            pw[(r + hi * 8) * 32 + 16 + nl] = p1[r];
        }
        __builtin_amdgcn_wave_barrier();

        // O += P(16x32) * V(32x128)
        for (int c = 0; c < 8; ++c) {
            int kk = c * 4 + khi;                     // even
            v2f ap = *(const v2f*)&pw[nl * 32 + kk];
            const float* vb = &Vs[(kk >> 1) * 256];
            for (int t = 0; t < 8; ++t) {
                v2f bv = *(const v2f*)&vb[(t * 16 + nl) * 2];
                o[t] = WMMA_F32(ap, bv, o[t]);
            }
        }
    }

    // normalize and write out (B,S,H,128)
    for (int r = 0; r < 8; ++r) {
        int row = q0 + w * 16 + r + hi * 8;
        float inv_l = 1.0f / lrow[r];
        float* op = O + ((size_t)(b * Sl + row) * 16 + h) * 128;
        for (int t = 0; t < 8; ++t)
            op[t * 16 + nl] = o[t][r] * inv_l;
    }
}

// ---------------------------------------------------------------------------
extern "C" void kernel_launch(void* const* d_in, const int* in_sizes, int n_in,
                              void* d_out, int out_size, void* d_ws, size_t ws_size,
                              hipStream_t stream)
{
    (void)in_sizes; (void)n_in; (void)out_size; (void)ws_size;
    const float* h_in  = (const float*)d_in[0];
    const float* W_dkv = (const float*)d_in[1];
    const float* b_dkv = (const float*)d_in[2];
    const float* W_dq  = (const float*)d_in[3];
    const float* b_dq  = (const float*)d_in[4];
    const float* W_uk  = (const float*)d_in[5];
    const float* b_uk  = (const float*)d_in[6];
    const float* W_uv  = (const float*)d_in[7];
    const float* b_uv  = (const float*)d_in[8];
    const float* W_uq  = (const float*)d_in[9];
    const float* b_uq  = (const float*)d_in[10];
    const float* W_qr  = (const float*)d_in[11];
    const float* b_qr  = (const float*)d_in[12];
    const float* W_kr  = (const float*)d_in[13];
    const float* b_kr  = (const float*)d_in[14];
    const float* W_out = (const float*)d_in[15];
    const float* b_out = (const float*)d_in[16];

    const int M = 4096;     // B*S
    const int S = 2048, H = 16, B = 2;

    float* out  = (float*)d_out;                       // (B,S,2048)
    float* ckv  = out + (size_t)M * 2048;              // (B,S,512)  (output #2)
    float* krot = ckv + (size_t)M * 512;               // (B,S,H,32) (output #3)

    float* ws = (float*)d_ws;
    float* cq = ws;                                    // M*512
    float* qf = cq + (size_t)M * 512;                  // M*2048 (B,S,H,128)
    float* kf = qf + (size_t)M * 2048;                 // M*2048
    float* vf = kf + (size_t)M * 2048;                 // M*2048
    float* ao = vf + (size_t)M * 2048;                 // M*2048 attn out

    dim3 blk(128);
    auto gemm = [&](const float* A, const float* W, const float* bias, float* o,
                    int K, int N, int ldo, int group, int hs, int off) {
        dim3 grid(M / 128, N / 64);
        mla_gemm_bias_kernel<<<grid, blk, 0, stream>>>(A, W, bias, o, M, K, N,
                                                       ldo, group, hs, off);
    };

    gemm(h_in, W_dkv, b_dkv, ckv, 2048, 512, 512, 512, 512, 0);     // c_kv
    gemm(h_in, W_dq,  b_dq,  cq,  2048, 512, 512, 512, 512, 0);     // c_q
    gemm(h_in, W_kr,  b_kr,  kf,  2048, 512, 2048, 32, 128, 96);    // k_rot pre
    gemm(ckv,  W_uk,  b_uk,  kf,  512, 1536, 2048, 96, 128, 0);     // k base
    gemm(ckv,  W_uv,  b_uv,  vf,  512, 2048, 2048, 2048, 2048, 0);  // v
    gemm(cq,   W_uq,  b_uq,  qf,  512, 1536, 2048, 96, 128, 0);     // q base
    gemm(cq,   W_qr,  b_qr,  qf,  512, 512,  2048, 32, 128, 96);    // q_rot pre

    int total = M * H;
    mla_rope_kernel<<<(total + 255) / 256, 256, 0, stream>>>(qf, nullptr, total, S);
    mla_rope_kernel<<<(total + 255) / 256, 256, 0, stream>>>(kf, krot, total, S);

    dim3 agrid(S / 32, H, B);
    mla_attn_kernel<<<agrid, 64, 0, stream>>>(qf, kf, vf, ao, S);

    gemm(ao, W_out, b_out, out, 2048, 2048, 2048, 2048, 2048, 0);   // final proj
}